// VectorQuantize_58428735094924
// MI455X (gfx1250) — compile-verified
//
#include <hip/hip_runtime.h>

typedef float v2f __attribute__((ext_vector_type(2)));
typedef float v8f __attribute__((ext_vector_type(8)));

constexpr int kD = 256;        // feature dim
constexpr int kC = 1024;       // codebook size
constexpr int kN = 64 * 512;   // B*T rows
constexpr int kLSTR = 260;     // padded LDS row stride (floats) -> bank-conflict free
constexpr float kDECAY = 0.99f;
constexpr float kEPS = 1e-5f;

// dynamic LDS layout (floats): xs[32*260] | es0[16*260] | es1[16*260]
constexpr int kXS_F = 32 * kLSTR;
constexpr int kES_F = 16 * kLSTR;
constexpr int kSMEM_BYTES = (kXS_F + 2 * kES_F) * 4;  // 66,560 B

__device__ __forceinline__ void async_ld_b128(unsigned lds_addr, const float* gaddr) {
  // GLOBAL_LOAD_ASYNC_TO_LDS_B128 (GV mode): copies 16 bytes from global gaddr to LDS address lds_addr, increments ASYNCcnt
  asm volatile("global_load_async_to_lds_b128 %0, %1, off"
               :: "v"(lds_addr), "v"(gaddr) : "memory");
}
__device__ __forceinline__ void wait_async_le16() {
  asm volatile("s_wait_asynccnt 0x10" ::: "memory");
}
__device__ __forceinline__ void wait_async_0() {
  asm volatile("s_wait_asynccnt 0x0" ::: "memory");
}

__global__ __launch_bounds__(256) void vq_zero(float* p, int n) {
  int i = blockIdx.x * blockDim.x + threadIdx.x;
  int stride = gridDim.x * blockDim.x;
  for (; i < n; i += stride) p[i] = 0.0f;
}

// norms[c] = ||embed[c]||^2   (one wave per code)
__global__ __launch_bounds__(32) void vq_norms(const float* __restrict__ embed,
                                               float* __restrict__ norms) {
  const int c = blockIdx.x;
  const int lane = threadIdx.x;
  const float* e = embed + (size_t)c * kD;
  float s = 0.0f;
#pragma unroll
  for (int j = 0; j < kD / 32; ++j) {
    float v = e[lane + 32 * j];
    s += v * v;
  }
#pragma unroll
  for (int off = 16; off >= 1; off >>= 1) s += __shfl_xor(s, off, 32);
  if (lane == 0) norms[c] = s;
}

// Main: per-block 32 rows; stream all 1024 codes in 16-wide tiles through
// v_wmma_f32_16x16x4_f32 with async double-buffered LDS staging; running
// argmax in registers; gather/scatter epilogue.
__global__ __launch_bounds__(64) void vq_main(const float* __restrict__ x,
                                              const float* __restrict__ embed,
                                              const float* __restrict__ norms,
                                              float* __restrict__ quant,
                                              float* __restrict__ ind_out,
                                              float* __restrict__ counts,
                                              float* __restrict__ esum) {
  extern __shared__ float smem[];
  float* xs = smem;                 // 32 x 260
  float* es0 = smem + kXS_F;        // 16 x 260 (buffer 0)
  float* es1 = es0 + kES_F;         // 16 x 260 (buffer 1)

  const int tid = threadIdx.x;
  const int wave = tid >> 5;
  const int lane = tid & 31;
  const int row0 = blockIdx.x * 32;

  const unsigned xs_lds = (unsigned)(size_t)xs;     // LDS byte address (low 32 bits of flat)
  const unsigned es_lds[2] = {(unsigned)(size_t)es0, (unsigned)(size_t)es1};

  // ---- async stage of the 32x256 x-tile (32 b128 per wave, FIFO-drained by first wait)
#pragma unroll
  for (int j = 0; j < 32; ++j) {
    int i = tid + 64 * j;
    int r = i >> 6;   // 64 float4 per row
    int c4 = i & 63;
    async_ld_b128(xs_lds + (unsigned)((r * kLSTR + c4 * 4) * 4),
                  x + (size_t)(row0 + r) * kD + c4 * 4);
  }
  // ---- async stage of embed tile 0 into buffer 0 (16 b128 per wave)
#pragma unroll
  for (int j = 0; j < 16; ++j) {
    int i = tid + 64 * j;
    int r = i >> 6;
    int c4 = i & 63;
    async_ld_b128(es_lds[0] + (unsigned)((r * kLSTR + c4 * 4) * 4),
                  embed + (size_t)r * kD + c4 * 4);
  }

  float bestv[8];
  int besti[8];
#pragma unroll
  for (int r = 0; r < 8; ++r) { bestv[r] = -3.402823466e38f; besti[r] = 0; }

  const int halfl = lane & 15;
  const int mrow = halfl + 16 * wave;   // A fragment row (per wave)
  const int koff = 2 * (lane >> 4);     // A/B fragment K sub-offset

  for (int ct = 0; ct < kC / 16; ++ct) {
    const int nb = (ct + 1) & 1;
    if (ct + 1 < kC / 16) {
      // prefetch next tile into the buffer last read in the previous iteration
      // (safe: all waves passed the trailing barrier of iteration ct-1)
#pragma unroll
      for (int j = 0; j < 16; ++j) {
        int i = tid + 64 * j;
        int r = i >> 6;
        int c4 = i & 63;
        async_ld_b128(es_lds[nb] + (unsigned)((r * kLSTR + c4 * 4) * 4),
                      embed + (size_t)((ct + 1) * 16 + r) * kD + c4 * 4);
      }
      wait_async_le16();  // tile ct (and, first time, the x tile) has landed
    } else {
      wait_async_0();
    }
    __syncthreads();

    const float* esb = (ct & 1) ? es1 : es0;
    v8f acc = {0.f, 0.f, 0.f, 0.f, 0.f, 0.f, 0.f, 0.f};
#pragma unroll 16
    for (int k0 = 0; k0 < kD; k0 += 4) {
      // A: 16x4 (M=lane&15, K = koff..koff+1), B: 4x16 (N=lane&15, K = koff..koff+1)
      v2f a = *reinterpret_cast<const v2f*>(&xs[mrow * kLSTR + k0 + koff]);
      v2f b = *reinterpret_cast<const v2f*>(&esb[halfl * kLSTR + k0 + koff]);
      acc = __builtin_amdgcn_wmma_f32_16x16x4_f32(false, a, false, b, (short)0,
                                                  acc, false, false);
    }

    const float nrm = norms[ct * 16 + halfl];
    const int baseidx = ct * 16 + halfl;
#pragma unroll
    for (int r = 0; r < 8; ++r) {
      // C/D layout: VGPR r <-> row r (lanes 0-15) / row r+8 (lanes 16-31), N = lane&15
      float v = 2.0f * acc[r] - nrm;
      int idx = baseidx;
#pragma unroll
      for (int off = 1; off < 16; off <<= 1) {  // reduce within 16-lane half
        float ov = __shfl_xor(v, off, 32);
        int oi = __shfl_xor(idx, off, 32);
        if (ov > v || (ov == v && oi < idx)) { v = ov; idx = oi; }
      }
      if (v > bestv[r]) { bestv[r] = v; besti[r] = idx; }  // tie -> keep earlier code
    }
    __syncthreads();  // everyone done reading buf[ct&1] before it is re-targeted
  }

  // Epilogue: indices, counts, quantize gather, embed_sum scatter
  const int wrow0 = row0 + 16 * wave;
#pragma unroll
  for (int m = 0; m < 16; ++m) {
    const int reg = m & 7;
    const int srcl = (m < 8) ? 0 : 16;
    const int c = __shfl(besti[reg], srcl, 32);
    const int grow = wrow0 + m;
    if (lane == srcl) {
      ind_out[grow] = (float)c;
      atomicAdd(&counts[c], 1.0f);
    }
    const float* ec = embed + (size_t)c * kD;
    float* qrow = quant + (size_t)grow * kD;
    float* er = esum + (size_t)c * kD;
    const int lrow = 16 * wave + m;
    for (int d = lane; d < kD; d += 32) {
      qrow[d] = ec[d];
      atomicAdd(&er[d], xs[lrow * kLSTR + d]);
    }
  }
}

// EMA update + Laplace smoothing + normalize (single block)
__global__ __launch_bounds__(1024) void vq_finalize(const float* __restrict__ embed_avg,
                                                    const float* __restrict__ cluster_size,
                                                    const float* __restrict__ counts,
                                                    const float* __restrict__ esum,
                                                    float* __restrict__ cs_out,
                                                    float* __restrict__ enorm_out) {
  __shared__ float sm[kC];
  __shared__ float red[32];
  const int t = threadIdx.x;
  const float cs = kDECAY * cluster_size[t] + (1.0f - kDECAY) * counts[t];
  cs_out[t] = cs;

  float s = cs;
#pragma unroll
  for (int off = 16; off >= 1; off >>= 1) s += __shfl_xor(s, off, 32);
  if ((t & 31) == 0) red[t >> 5] = s;
  __syncthreads();
  if (t < 32) {
    float v = red[t];
#pragma unroll
    for (int off = 16; off >= 1; off >>= 1) v += __shfl_xor(v, off, 32);
    if (t == 0) red[0] = v;
  }
  __syncthreads();
  const float n = red[0];
  sm[t] = (cs + kEPS) / (n + (float)kC * kEPS) * n;
  __syncthreads();

  for (int i = t; i < kC * kD; i += 1024) {
    int cc = i >> 8;  // kD == 256
    enorm_out[i] = (kDECAY * embed_avg[i] + (1.0f - kDECAY) * esum[i]) / sm[cc];
  }
}

extern "C" void kernel_launch(void* const* d_in, const int* in_sizes, int n_in,
                              void* d_out, int out_size, void* d_ws, size_t ws_size,
                              hipStream_t stream) {
  const float* x = (const float*)d_in[0];             // [64,512,256]
  const float* embed = (const float*)d_in[1];         // [1,1024,256]
  const float* embed_avg = (const float*)d_in[2];     // [1,1024,256]
  const float* cluster_size = (const float*)d_in[3];  // [1,1024]

  float* out = (float*)d_out;
  float* quant = out;                              // N*D
  float* ind = quant + (size_t)kN * kD;            // N
  float* enorm = ind + kN;                         // C*D
  float* cs = enorm + (size_t)kC * kD;             // C

  float* wsf = (float*)d_ws;
  float* counts = wsf;                   // C
  float* esum = wsf + kC;                // C*D
  float* norms = esum + (size_t)kC * kD; // C

  vq_zero<<<512, 256, 0, stream>>>(wsf, kC + kC * kD);
  vq_norms<<<kC, 32, 0, stream>>>(embed, norms);
  vq_main<<<kN / 32, 64, kSMEM_BYTES, stream>>>(x, embed, norms, quant, ind, counts, esum);
  vq_finalize<<<1, 1024, 0, stream>>>(embed_avg, cluster_size, counts, esum, cs, enorm);
}